// MS_MSA_73839077753033
// MI455X (gfx1250) — compile-verified
//
#include <hip/hip_runtime.h>

typedef float v2f __attribute__((ext_vector_type(2)));
typedef float v8f __attribute__((ext_vector_type(8)));

#define HEADS   4
#define Cc      64
#define Nn      262144      // 512*512
#define Bb      2
#define EPSV    1e-12f

// ---- gram pass config ----
#define G_WGS   128         // workgroups per batch
#define G_TILE  64          // columns per LDS tile (double buffered)
#define G_ITERS 32          // 128 WG * 32 tiles * 64 cols = 262144
#define G_PITCH 68          // 68 % 64 == 4 -> conflict-free b64 lane pattern

// ---- apply pass config ----
#define K3_WGS   256        // workgroups per batch
#define K3_ITERS 8          // 256 WG * 8 strips * 128 cols = 262144
#define M_PITCH  68         // 68 % 64 == 4 -> conflict-free

static __device__ __forceinline__ v8f wmma_f32(v2f a, v2f b, v8f c) {
    return __builtin_amdgcn_wmma_f32_16x16x4_f32(false, a, false, b, (short)0, c,
                                                 false, false);
}

// ============================================================================
// Kernel 1: partial Gram matrices  G_part[b][wg][64][64] = X_tile * X_tile^T
// Double-buffered via GLOBAL_LOAD_ASYNC_TO_LDS_B128 (ASYNCcnt tracked DMA).
// ============================================================================
__global__ __launch_bounds__(256) void gram_kernel(const float* __restrict__ x,
                                                   float* __restrict__ part) {
    __shared__ float tile[2][Cc * G_PITCH];   // 2 x 17408 B

    const int b    = blockIdx.y;
    const int wg   = blockIdx.x;
    const int tid  = threadIdx.x;
    const int lane = tid & 31;
    const int wave = tid >> 5;
    const int lm   = lane & 15;
    const int kh   = (lane >> 4) << 1;   // 0 for lanes 0-15, 2 for lanes 16-31

    // wave owns two of the 16 (bi,bj) 16x16 sub-tiles of G
    const int p0 = wave * 2, p1 = p0 + 1;
    const int bi0 = p0 >> 2, bj0 = p0 & 3;
    const int bi1 = p1 >> 2, bj1 = p1 & 3;

    const float* xb = x + (size_t)b * Cc * Nn;

    const uint32_t lbase    = (uint32_t)(uintptr_t)&tile[0][0];
    const uint32_t bufbytes = Cc * G_PITCH * 4u;

    // issue one tile's cooperative load: 64 rows x 64 cols, b128 per lane,
    // 4 async-DMA ops per thread (ASYNCcnt += 4 per wave)
    auto issue_tile = [&](int buf, int t) {
        const int n0 = (wg * G_ITERS + t) * G_TILE;
        for (int i = 0; i < 4; ++i) {
            const int linear = i * 256 + tid;      // float4 index, 16 per row
            const int row    = linear >> 4;
            const int c4     = linear & 15;
            const float*  g = xb + (size_t)row * Nn + n0 + c4 * 4;
            const uint32_t l = lbase + (uint32_t)buf * bufbytes +
                               (uint32_t)(row * G_PITCH + c4 * 4) * 4u;
            asm volatile("global_load_async_to_lds_b128 %0, %1, off"
                         :: "v"(l), "v"(g) : "memory");
        }
    };

    v8f acc0 = {};
    v8f acc1 = {};

    issue_tile(0, 0);

    for (int t = 0; t < G_ITERS; ++t) {
        if (t + 1 < G_ITERS) {
            issue_tile((t + 1) & 1, t + 1);       // prefetch next tile
            // two 4-op groups in flight; async ops retire in order, so
            // waiting <=4 guarantees the OLDER tile's group is complete
            asm volatile("s_wait_asynccnt 4" ::: "memory");
        } else {
            asm volatile("s_wait_asynccnt 0" ::: "memory");
        }
        __syncthreads();                          // tile visible to all waves

        const float* tb = &tile[t & 1][0];
        for (int ks = 0; ks < G_TILE / 4; ++ks) {
            const int kb = ks * 4 + kh;
            const float* ra0 = &tb[(bi0 * 16 + lm) * G_PITCH + kb];
            const float* rb0 = &tb[(bj0 * 16 + lm) * G_PITCH + kb];
            const float* ra1 = &tb[(bi1 * 16 + lm) * G_PITCH + kb];
            const float* rb1 = &tb[(bj1 * 16 + lm) * G_PITCH + kb];
            v2f a0 = {ra0[0], ra0[1]};
            v2f b0 = {rb0[0], rb0[1]};
            v2f a1 = {ra1[0], ra1[1]};
            v2f b1 = {rb1[0], rb1[1]};
            acc0 = wmma_f32(a0, b0, acc0);
            acc1 = wmma_f32(a1, b1, acc1);
        }
        __syncthreads();                          // buffer safe to overwrite
    }

    // write partial G (deterministic, no atomics)
    float* pw = part + (size_t)(b * G_WGS + wg) * (Cc * Cc);
    const int rofs = (lane >> 4) << 3;  // 0 or 8
    for (int r = 0; r < 8; ++r) {
        pw[(bi0 * 16 + r + rofs) * Cc + bj0 * 16 + lm] = acc0[r];
        pw[(bi1 * 16 + r + rofs) * Cc + bj1 * 16 + lm] = acc1[r];
    }
}

// ============================================================================
// Kernel 2: reduce partials -> G; tiny attention math -> M = Wp*A*Wv + I
// ============================================================================
__global__ __launch_bounds__(256) void attn_kernel(const float* __restrict__ part,
                                                   const float* __restrict__ Wq,
                                                   const float* __restrict__ Wk,
                                                   const float* __restrict__ Wv,
                                                   const float* __restrict__ Wp,
                                                   const float* __restrict__ rescale,
                                                   float* __restrict__ Mout) {
    __shared__ float G[Cc * Cc];           // 16 KB
    __shared__ float A1[HEADS][Cc][16];    // 16 KB  (G * Wq_h^T)
    __shared__ float S[HEADS][16][16];     // 4 KB   (raw scores -> softmax)
    __shared__ float qn[HEADS][16];
    __shared__ float kn[HEADS][16];
    __shared__ float T[Cc][Cc + 1];        // ~16.3 KB (blockdiag(attn) * Wv)

    const int tid = threadIdx.x;
    const int h   = tid >> 6;       // 4 head-groups of 64 threads
    const int t64 = tid & 63;

    for (int b = 0; b < Bb; ++b) {
        // ---- reduce partial Grams ----
        for (int i = tid; i < Cc * Cc; i += 256) {
            float s = 0.f;
            for (int w = 0; w < G_WGS; ++w)
                s += part[(size_t)(b * G_WGS + w) * (Cc * Cc) + i];
            G[i] = s;
        }
        __syncthreads();

        // ---- A1[c][e] = sum_c2 G[c][c2] * Wq_h[e][c2] ----
        for (int idx = t64; idx < Cc * 16; idx += 64) {
            const int c = idx >> 4, e = idx & 15;
            const float* wq = Wq + (h * 16 + e) * Cc;
            float s = 0.f;
            for (int c2 = 0; c2 < Cc; ++c2) s += G[c * Cc + c2] * wq[c2];
            A1[h][c][e] = s;
        }
        __syncthreads();

        // ---- scores S[d][e] = Wk_h[d] . A1[:,e];  norms ----
        for (int idx = t64; idx < 256; idx += 64) {
            const int dd = idx >> 4, e = idx & 15;
            const float* wk = Wk + (h * 16 + dd) * Cc;
            float s = 0.f;
            for (int c = 0; c < Cc; ++c) s += wk[c] * A1[h][c][e];
            S[h][dd][e] = s;
        }
        if (t64 < 16) {
            const int e = t64;
            const float* wq = Wq + (h * 16 + e) * Cc;
            float s = 0.f;
            for (int c = 0; c < Cc; ++c) s += wq[c] * A1[h][c][e];
            qn[h][e] = sqrtf(fmaxf(s, 0.f));
        } else if (t64 < 32) {
            const int dd = t64 - 16;
            const float* wk = Wk + (h * 16 + dd) * Cc;
            float s2 = 0.f;
            for (int c2 = 0; c2 < Cc; ++c2) {
                float t = 0.f;
                for (int c = 0; c < Cc; ++c) t += wk[c] * G[c * Cc + c2];
                s2 += t * wk[c2];
            }
            kn[h][dd] = sqrtf(fmaxf(s2, 0.f));
        }
        __syncthreads();

        // ---- normalize + rescale + softmax (row d over e) ----
        if (t64 < 16) {
            const int dd = t64;
            const float rs = rescale[h];
            const float kd = fmaxf(kn[h][dd], EPSV);
            float z[16], mx = -1e30f;
            for (int e = 0; e < 16; ++e) {
                const float v = S[h][dd][e] * rs / (kd * fmaxf(qn[h][e], EPSV));
                z[e] = v;
                mx = fmaxf(mx, v);
            }
            float sum = 0.f;
            for (int e = 0; e < 16; ++e) { z[e] = __expf(z[e] - mx); sum += z[e]; }
            const float inv = 1.f / sum;
            for (int e = 0; e < 16; ++e) S[h][dd][e] = z[e] * inv;
        }
        __syncthreads();

        // ---- T[16h+d][c] = sum_e attn[d][e] * Wv[16h+e][c] ----
        for (int idx = t64; idx < 16 * Cc; idx += 64) {
            const int dd = idx >> 6, c = idx & 63;
            float s = 0.f;
            for (int e = 0; e < 16; ++e) s += S[h][dd][e] * Wv[(h * 16 + e) * Cc + c];
            T[h * 16 + dd][c] = s;
        }
        __syncthreads();

        // ---- M = Wp * T  (+ I to fold residual) ----
        for (int idx = tid; idx < Cc * Cc; idx += 256) {
            const int o = idx >> 6, c = idx & 63;
            const float* wp = Wp + o * Cc;
            float s = 0.f;
            for (int r = 0; r < Cc; ++r) s += wp[r] * T[r][c];
            if (o == c) s += 1.f;
            Mout[b * Cc * Cc + idx] = s;
        }
        __syncthreads();
    }
}

// ============================================================================
// Kernel 3: out[b] = M[b] @ x[b] + bp   (residual already folded into M)
// ============================================================================
__global__ __launch_bounds__(256) void apply_kernel(const float* __restrict__ x,
                                                    const float* __restrict__ Mf,
                                                    const float* __restrict__ bp,
                                                    float* __restrict__ out) {
    __shared__ float Ml[Cc * M_PITCH];

    const int b    = blockIdx.y;
    const int tid  = threadIdx.x;
    const int lane = tid & 31;
    const int wave = tid >> 5;
    const int lm   = lane & 15;
    const int kh   = (lane >> 4) << 1;
    const int rofs = (lane >> 4) << 3;

    // stage M in LDS (coalesced), then preload all A operands into registers
    for (int i = tid; i < Cc * Cc; i += 256)
        Ml[(i >> 6) * M_PITCH + (i & 63)] = Mf[b * Cc * Cc + i];
    __syncthreads();

    v2f A[4][16];
    for (int bi = 0; bi < 4; ++bi)
        for (int ks = 0; ks < 16; ++ks) {
            const float* p = &Ml[(bi * 16 + lm) * M_PITCH + ks * 4 + kh];
            A[bi][ks] = (v2f){p[0], p[1]};
        }

    const size_t boff = (size_t)b * Cc * Nn;
    const float* xb = x + boff;
    float* ob       = out + boff;

    for (int it = 0; it < K3_ITERS; ++it) {
        const int n0  = (blockIdx.x * K3_ITERS + it) * 128;
        const int col = n0 + wave * 16 + lm;     // wave owns one 16-col slab

        v8f acc0 = {}, acc1 = {}, acc2 = {}, acc3 = {};
        for (int ks = 0; ks < 16; ++ks) {
            const int kb = ks * 4 + kh;
            v2f bv = {xb[(size_t)kb * Nn + col], xb[(size_t)(kb + 1) * Nn + col]};
            acc0 = wmma_f32(A[0][ks], bv, acc0);
            acc1 = wmma_f32(A[1][ks], bv, acc1);
            acc2 = wmma_f32(A[2][ks], bv, acc2);
            acc3 = wmma_f32(A[3][ks], bv, acc3);
        }

        for (int r = 0; r < 8; ++r) {
            const int m = r + rofs;
            ob[(size_t)(0 * 16 + m) * Nn + col] = acc0[r] + bp[0 * 16 + m];
            ob[(size_t)(1 * 16 + m) * Nn + col] = acc1[r] + bp[1 * 16 + m];
            ob[(size_t)(2 * 16 + m) * Nn + col] = acc2[r] + bp[2 * 16 + m];
            ob[(size_t)(3 * 16 + m) * Nn + col] = acc3[r] + bp[3 * 16 + m];
        }
    }
}

// ============================================================================
extern "C" void kernel_launch(void* const* d_in, const int* in_sizes, int n_in,
                              void* d_out, int out_size, void* d_ws, size_t ws_size,
                              hipStream_t stream) {
    (void)in_sizes; (void)n_in; (void)out_size; (void)ws_size;

    const float* x       = (const float*)d_in[0];
    const float* Wq      = (const float*)d_in[1];
    const float* Wk      = (const float*)d_in[2];
    const float* Wv      = (const float*)d_in[3];
    const float* Wp      = (const float*)d_in[4];
    const float* bp      = (const float*)d_in[5];
    const float* rescale = (const float*)d_in[6];
    float* out = (float*)d_out;

    float* part = (float*)d_ws;                        // [Bb][G_WGS][64*64] = 4 MB
    float* Mf   = part + (size_t)Bb * G_WGS * Cc * Cc; // [Bb][64*64]       = 32 KB

    gram_kernel<<<dim3(G_WGS, Bb), 256, 0, stream>>>(x, part);
    attn_kernel<<<dim3(1, 1), 256, 0, stream>>>(part, Wq, Wk, Wv, Wp, rescale, Mf);
    apply_kernel<<<dim3(K3_WGS, Bb), 256, 0, stream>>>(x, Mf, bp, out);
}